// Decoder_14224931684806
// MI455X (gfx1250) — compile-verified
//
#include <hip/hip_runtime.h>
#include <hip/hip_bf16.h>

typedef float v2f __attribute__((ext_vector_type(2)));
typedef float v8f __attribute__((ext_vector_type(8)));
typedef int v4i_t __attribute__((vector_size(4 * sizeof(int))));

#define NEG_SLOPE 0.2f

#define AS1 __attribute__((address_space(1)))
#define AS3 __attribute__((address_space(3)))

#if __has_builtin(__builtin_amdgcn_global_load_async_to_lds_b128) && \
    __has_builtin(__builtin_amdgcn_s_wait_asynccnt)
#define USE_ASYNC_LDS 1
#else
#define USE_ASYNC_LDS 0
#endif

// ---------------------------------------------------------------------------
// Monotone float <-> uint encoding so we can use integer atomicMax for the
// edge-softmax running max (covers negative values correctly).
// ---------------------------------------------------------------------------
__device__ __forceinline__ unsigned f2ord(float f) {
  unsigned u = __float_as_uint(f);
  return (u & 0x80000000u) ? ~u : (u | 0x80000000u);
}
__device__ __forceinline__ float ord2f(unsigned u) {
  return (u & 0x80000000u) ? __uint_as_float(u & 0x7fffffffu)
                           : __uint_as_float(~u);
}

// ---------------------------------------------------------------------------
// fp32 WMMA GEMM: C[M,NC] = A[M,128] * B[128,NC], M % 16 == 0, NC % 128 == 0.
// Block = 256 threads = 8 waves; each wave computes one 16x16 tile with
// V_WMMA_F32_16X16X4_F32, K fully unrolled (immediate address offsets).
// A tile (16x128) is staged in LDS via async global->LDS copies; row stride
// 132 floats keeps 16B alignment per chunk and is bank-conflict free.
// ---------------------------------------------------------------------------
#define LDSK 132

template <int NC>
__global__ __launch_bounds__(256) void gemm_wmma(const float* __restrict__ A,
                                                 const float* __restrict__ B,
                                                 float* __restrict__ C) {
  __shared__ float lds[16 * LDSK];
  const int tid  = threadIdx.x;
  const int lane = tid & 31;
  const int wave = tid >> 5;
  const int mbase = blockIdx.x * 16;

  // ---- stage A tile [16 x 128] into LDS -----------------------------------
#if USE_ASYNC_LDS
  {
    int chunk = tid;  // 512 x 16B chunks, 2 per thread
#pragma unroll
    for (int it = 0; it < 2; ++it, chunk += 256) {
      int r = chunk >> 5;          // row 0..15
      int c = (chunk & 31) << 2;   // col 0..124 step 4
      __builtin_amdgcn_global_load_async_to_lds_b128(
          (AS1 v4i_t*)(A + (size_t)(mbase + r) * 128 + c),
          (AS3 v4i_t*)(&lds[r * LDSK + c]), 0, 0);
    }
    __builtin_amdgcn_s_wait_asynccnt(0);
  }
#else
  for (int j = tid; j < 16 * 128; j += 256) {
    int r = j >> 7, c = j & 127;
    lds[r * LDSK + c] = A[(size_t)(mbase + r) * 128 + c];
  }
#endif
  __syncthreads();

  const int nbase = blockIdx.y * 128 + wave * 16;
  const int ncol  = lane & 15;          // B/C column within tile
  const int mrow  = lane & 15;          // A row within tile
  const int khalf = (lane >> 4) << 1;   // 0 or 2 (K sub-offset, 16x4 layout)
  const int mhi   = (lane >> 4) << 3;   // 0 or 8 (C row group)

  const float* Bp = B + (size_t)khalf * NC + nbase + ncol;
  __builtin_prefetch(Bp, 0, 0);  // global_prefetch_b8 (in-bounds)

  v8f acc0 = {}, acc1 = {};
#pragma unroll
  for (int k = 0; k < 128; k += 8) {
    v2f a0, b0, a1, b1;
    a0.x = lds[mrow * LDSK + k + khalf];
    a0.y = lds[mrow * LDSK + k + khalf + 1];
    b0.x = Bp[(size_t)k * NC];
    b0.y = Bp[(size_t)(k + 1) * NC];
    a1.x = lds[mrow * LDSK + k + 4 + khalf];
    a1.y = lds[mrow * LDSK + k + 4 + khalf + 1];
    b1.x = Bp[(size_t)(k + 4) * NC];
    b1.y = Bp[(size_t)(k + 5) * NC];
    acc0 = __builtin_amdgcn_wmma_f32_16x16x4_f32(false, a0, false, b0,
                                                 (short)0, acc0, false, false);
    acc1 = __builtin_amdgcn_wmma_f32_16x16x4_f32(false, a1, false, b1,
                                                 (short)0, acc1, false, false);
  }
#pragma unroll
  for (int r = 0; r < 8; ++r) acc0[r] += acc1[r];

  float* cptr = C + (size_t)(mbase + mhi) * NC + nbase + ncol;
#pragma unroll
  for (int r = 0; r < 8; ++r) cptr[(size_t)r * NC] = acc0[r];
}

// ---------------------------------------------------------------------------
// Zero/reset the per-layer accumulation buffers.
// ---------------------------------------------------------------------------
__global__ void init_buffers(unsigned* maxb, float* denom, float* agg,
                             int nh, int na) {
  int i = blockIdx.x * blockDim.x + threadIdx.x;
  int stride = gridDim.x * blockDim.x;
  for (int j = i; j < na; j += stride) agg[j] = 0.0f;
  for (int j = i; j < nh; j += stride) {
    maxb[j] = 0u;        // minimum of the monotone encoding
    denom[j] = 0.0f;
  }
}

// el[n,h] = sum_d feat[n,h,d]*al[h,d];  er likewise with ar.  D == 64.
__global__ void attn_scores(const float* __restrict__ feat,
                            const float* __restrict__ al,
                            const float* __restrict__ ar,
                            float* __restrict__ el, float* __restrict__ er,
                            int N, int H) {
  int i = blockIdx.x * blockDim.x + threadIdx.x;
  if (i >= N * H) return;
  int h = i % H, n = i / H;
  const float* f = feat + ((size_t)n * H + h) * 64;
  const float* a = al + h * 64;
  const float* b = ar + h * 64;
  float sl = 0.0f, sr = 0.0f;
#pragma unroll
  for (int d = 0; d < 64; d += 4) {
    float4 fv = *(const float4*)(f + d);
    float4 av = *(const float4*)(a + d);
    float4 bv = *(const float4*)(b + d);
    sl += fv.x * av.x + fv.y * av.y + fv.z * av.z + fv.w * av.w;
    sr += fv.x * bv.x + fv.y * bv.y + fv.z * bv.z + fv.w * bv.w;
  }
  el[i] = sl;
  er[i] = sr;
}

__device__ __forceinline__ float edge_score(const float* el, const float* er,
                                            int s, int d, int h, int H) {
  float v = el[s * H + h] + er[d * H + h];
  return v > 0.0f ? v : NEG_SLOPE * v;   // leaky_relu
}

__global__ void edge_max(const int* __restrict__ src, const int* __restrict__ dst,
                         const float* __restrict__ el, const float* __restrict__ er,
                         unsigned* __restrict__ maxb, int EH, int H) {
  int i = blockIdx.x * blockDim.x + threadIdx.x;
  if (i >= EH) return;
  int h = i % H, e = i / H;
  int s = src[e], d = dst[e];
  float v = edge_score(el, er, s, d, h, H);
  atomicMax(&maxb[d * H + h], f2ord(v));
}

__global__ void edge_exp(const int* __restrict__ src, const int* __restrict__ dst,
                         const float* __restrict__ el, const float* __restrict__ er,
                         const unsigned* __restrict__ maxb,
                         float* __restrict__ denom, float* __restrict__ exbuf,
                         int EH, int H) {
  int i = blockIdx.x * blockDim.x + threadIdx.x;
  if (i >= EH) return;
  int h = i % H, e = i / H;
  int s = src[e], d = dst[e];
  float v = edge_score(el, er, s, d, h, H);
  float m = ord2f(maxb[d * H + h]);
  float ex = __expf(v - m);
  exbuf[i] = ex;
  atomicAdd(&denom[d * H + h], ex);
}

// msg = feat[src,h,:] * alpha ; agg[dst,h,:] += msg.
// 16 threads per (edge, head); each handles 4 consecutive d via float4.
__global__ void edge_agg(const int* __restrict__ src, const int* __restrict__ dst,
                         const float* __restrict__ exbuf,
                         const float* __restrict__ denom,
                         const float* __restrict__ feat,
                         float* __restrict__ agg, int E, int H) {
  int i = blockIdx.x * blockDim.x + threadIdx.x;
  if (i >= E * H * 16) return;
  int dq = i & 15;
  int t = i >> 4;
  int h = t % H;
  int e = t / H;
  int s = src[e], d = dst[e];
  float alpha = exbuf[(size_t)e * H + h] / denom[d * H + h];
  const float4 f = *(const float4*)(feat + ((size_t)s * H + h) * 64 + dq * 4);
  float* o = agg + ((size_t)d * H + h) * 64 + dq * 4;
  atomicAdd(o + 0, f.x * alpha);
  atomicAdd(o + 1, f.y * alpha);
  atomicAdd(o + 2, f.z * alpha);
  atomicAdd(o + 3, f.w * alpha);
}

// h = elu(agg + bias),  elu(x) = x>0 ? x : exp(x)-1
__global__ void finalize_elu(const float* __restrict__ agg,
                             const float* __restrict__ bias,
                             float* __restrict__ h, int total, int C) {
  int i = blockIdx.x * blockDim.x + threadIdx.x;
  if (i >= total) return;
  float v = agg[i] + bias[i % C];
  h[i] = v > 0.0f ? v : (__expf(v) - 1.0f);
}

// out[n,d] = mean_h relu(agg[n,h,d] + b2[h,d]),  H==4, D==64
__global__ void finalize_out(const float* __restrict__ agg,
                             const float* __restrict__ bias,
                             float* __restrict__ out, int N) {
  int i = blockIdx.x * blockDim.x + threadIdx.x;
  if (i >= N * 64) return;
  int d = i & 63;
  int n = i >> 6;
  float s = 0.0f;
#pragma unroll
  for (int h = 0; h < 4; ++h) {
    float v = agg[((size_t)n * 4 + h) * 64 + d] + bias[h * 64 + d];
    s += v > 0.0f ? v : 0.0f;
  }
  out[i] = 0.25f * s;
}

// ---------------------------------------------------------------------------
extern "C" void kernel_launch(void* const* d_in, const int* in_sizes, int n_in,
                              void* d_out, int out_size, void* d_ws, size_t ws_size,
                              hipStream_t stream) {
  (void)in_sizes; (void)n_in; (void)out_size; (void)ws_size;
  const float* feature = (const float*)d_in[0];
  const int*   src     = (const int*)d_in[1];
  const int*   dst     = (const int*)d_in[2];
  const float* W1      = (const float*)d_in[3];
  const float* al1     = (const float*)d_in[4];
  const float* ar1     = (const float*)d_in[5];
  const float* b1      = (const float*)d_in[6];
  const float* W2      = (const float*)d_in[7];
  const float* al2     = (const float*)d_in[8];
  const float* ar2     = (const float*)d_in[9];
  const float* b2      = (const float*)d_in[10];
  float* out = (float*)d_out;

  const int N = 50000, E = 800000;

  float* ws    = (float*)d_ws;
  float* feat  = ws;                                  // N*256 (max of layers)
  float* agg   = feat + (size_t)N * 256;              // N*256
  float* hbuf  = agg + (size_t)N * 256;               // N*128
  float* el    = hbuf + (size_t)N * 128;              // N*4
  float* er    = el + (size_t)N * 4;                  // N*4
  unsigned* maxb = (unsigned*)(er + (size_t)N * 4);   // N*4
  float* denom = (float*)(maxb + (size_t)N * 4);      // N*4
  float* exbuf = denom + (size_t)N * 4;               // E*4

  const int TB = 256;
  // ------------------------- Layer 1: H=2, D=64 ----------------------------
  {
    const int H = 2;
    const int C = H * 64;                 // 128
    init_buffers<<<(N * C + TB - 1) / TB, TB, 0, stream>>>(maxb, denom, agg,
                                                           N * H, N * C);
    gemm_wmma<128><<<dim3(N / 16, 1), TB, 0, stream>>>(feature, W1, feat);
    attn_scores<<<(N * H + TB - 1) / TB, TB, 0, stream>>>(feat, al1, ar1, el, er, N, H);
    edge_max<<<(E * H + TB - 1) / TB, TB, 0, stream>>>(src, dst, el, er, maxb, E * H, H);
    edge_exp<<<(E * H + TB - 1) / TB, TB, 0, stream>>>(src, dst, el, er, maxb,
                                                       denom, exbuf, E * H, H);
    edge_agg<<<(E * H * 16 + TB - 1) / TB, TB, 0, stream>>>(src, dst, exbuf, denom,
                                                            feat, agg, E, H);
    finalize_elu<<<(N * C + TB - 1) / TB, TB, 0, stream>>>(agg, b1, hbuf, N * C, C);
  }
  // ------------------------- Layer 2: H=4, D=64 ----------------------------
  {
    const int H = 4;
    const int C = H * 64;                 // 256
    init_buffers<<<(N * C + TB - 1) / TB, TB, 0, stream>>>(maxb, denom, agg,
                                                           N * H, N * C);
    gemm_wmma<256><<<dim3(N / 16, 2), TB, 0, stream>>>(hbuf, W2, feat);
    attn_scores<<<(N * H + TB - 1) / TB, TB, 0, stream>>>(feat, al2, ar2, el, er, N, H);
    edge_max<<<(E * H + TB - 1) / TB, TB, 0, stream>>>(src, dst, el, er, maxb, E * H, H);
    edge_exp<<<(E * H + TB - 1) / TB, TB, 0, stream>>>(src, dst, el, er, maxb,
                                                       denom, exbuf, E * H, H);
    edge_agg<<<(E * H * 16 + TB - 1) / TB, TB, 0, stream>>>(src, dst, exbuf, denom,
                                                            feat, agg, E, H);
    finalize_out<<<(N * 64 + TB - 1) / TB, TB, 0, stream>>>(agg, b2, out, N);
  }
}